// MoEReduceRSTensorParallel_50672024158953
// MI455X (gfx1250) — compile-verified
//
#include <hip/hip_runtime.h>
#include <hip/hip_bf16.h>
#include <stdint.h>

// ---------------- problem constants ----------------
#define T_    1024
#define K_    2
#define E_    128
#define I_    512
#define H_    2048
#define R_    (T_*K_)          // 2048 rows

// ---------------- tiling constants -----------------
#define HTILE 128              // H columns per block (8 waves * 16)
#define KC    32               // K-chunk of weights staged in LDS
#define MSUP  32               // rows (2 m-groups of 16) per pass
#define WSTR  (HTILE + 4)      // 132 floats: LDS row stride for weights
#define XSTR  (I_ + 4)         // 516 floats: LDS row stride for x (full-K resident)

typedef __attribute__((ext_vector_type(2))) float v2f;
typedef __attribute__((ext_vector_type(4))) float v4f;
typedef __attribute__((ext_vector_type(8))) float v8f;

// ---------------------------------------------------------------------------
// Kernel 1: bucket rows by expert. counts[e], offsets[e], rowlist[R_] in ws.
// ---------------------------------------------------------------------------
__global__ __launch_bounds__(256) void moe_bin(const int* __restrict__ ce,
                                               int* __restrict__ counts,
                                               int* __restrict__ offsets,
                                               int* __restrict__ rowlist) {
    __shared__ int cnt[E_];
    __shared__ int cur[E_];
    __shared__ int off[E_ + 1];
    const int t = threadIdx.x;
    if (t < E_) cnt[t] = 0;
    __syncthreads();
    for (int r = t; r < R_; r += 256) atomicAdd(&cnt[ce[r]], 1);
    __syncthreads();
    if (t == 0) {
        int s = 0;
        for (int i = 0; i < E_; ++i) { off[i] = s; s += cnt[i]; }
        off[E_] = s;
    }
    __syncthreads();
    if (t < E_) cur[t] = off[t];
    __syncthreads();
    for (int r = t; r < R_; r += 256) {
        int e = ce[r];
        int p = atomicAdd(&cur[e], 1);
        rowlist[p] = r;
    }
    if (t < E_) { counts[t] = cnt[t]; offsets[t] = off[t]; }
}

// ---------------------------------------------------------------------------
// Async weight-chunk stage: KC x HTILE fp32 (16 KB) straight into LDS via
// GLOBAL_LOAD_ASYNC_TO_LDS_B128 (tracked with ASYNCcnt). 4 instr per wave.
// ---------------------------------------------------------------------------
__device__ __forceinline__ void wchunk_async(float* dst /*LDS*/,
                                             const float* __restrict__ src,
                                             int t, int kc) {
#pragma unroll
    for (int it = 0; it < (KC * HTILE) / (256 * 4); ++it) {   // 4 iters
        int f  = it * 256 + t;
        int kr = f >> 5;                 // 32 float4 per k-row
        int cv = (f & 31) * 4;
        const float* gp = src + (size_t)(kc + kr) * H_ + cv;
        unsigned lo = (unsigned)(uintptr_t)(dst + kr * WSTR + cv); // LDS byte offset
        asm volatile("global_load_async_to_lds_b128 %0, %1, off"
                     :: "v"(lo), "v"(gp) : "memory");
    }
}

__device__ __forceinline__ void wait_async0() {
    asm volatile("s_wait_asynccnt 0x0" ::: "memory");
}

// ---------------------------------------------------------------------------
// Branch-free K loop specialized on the number of active 16-row m-groups.
// Double-buffered LDS weight chunks overlapped with WMMA compute.
// ---------------------------------------------------------------------------
template<int NMG>
__device__ __forceinline__ void gemm_pass(const float* __restrict__ wg,
                                          float (*wl)[KC * WSTR],
                                          const float* __restrict__ xl,
                                          int t, int l15, int koff, int ncol,
                                          v8f* acc) {
    // preload chunk 0
    wchunk_async(wl[0], wg, t, 0);
    wait_async0();
    __syncthreads();
#pragma unroll 1
    for (int kc = 0; kc < I_; kc += KC) {
        const int cur = (kc / KC) & 1;
        if (kc + KC < I_) wchunk_async(wl[cur ^ 1], wg, t, kc + KC);  // prefetch next
        const float* wlc = wl[cur];
        const float* xlc = xl + kc;
#pragma unroll
        for (int kk = 0; kk < KC; kk += 4) {
            const int kb = kk + koff;
            v2f b;
            b.x = wlc[(kb + 0) * WSTR + ncol];   // B[k  ][n]
            b.y = wlc[(kb + 1) * WSTR + ncol];   // B[k+1][n]
#pragma unroll
            for (int mg = 0; mg < NMG; ++mg) {
                v2f a = *(const v2f*)&xlc[(mg * 16 + l15) * XSTR + kb];
                acc[mg] = __builtin_amdgcn_wmma_f32_16x16x4_f32(
                    false, a, false, b, (short)0, acc[mg], false, false);
            }
        }
        wait_async0();        // next chunk has landed (overlapped with compute)
        __syncthreads();
    }
}

// ---------------------------------------------------------------------------
// Kernel 2: grouped GEMM via V_WMMA_F32_16X16X4_F32.
// Block = (expert, 128-wide H tile); 8 waves, each wave owns 16 N columns.
// ---------------------------------------------------------------------------
__global__ __launch_bounds__(256) void moe_gemm(const float* __restrict__ x,
                                                const float* __restrict__ w,
                                                const int* __restrict__ counts,
                                                const int* __restrict__ offsets,
                                                const int* __restrict__ rowlist,
                                                float* __restrict__ y) {
    __shared__ float wl[2][KC * WSTR];   // 2 x 16.5 KB weight chunks
    __shared__ float xl[MSUP * XSTR];    // 32 rows x full K=512 (~66 KB)
    __shared__ int   rl[MSUP];

    const int e = blockIdx.x;
    const int cnt = counts[e];
    if (cnt == 0) return;
    const int off   = offsets[e];
    const int hbase = blockIdx.y * HTILE;

    const int t     = threadIdx.x;
    const int lane  = t & 31;
    const int wave  = t >> 5;
    const int l15   = lane & 15;
    const int lhalf = lane >> 4;
    const int koff  = lhalf * 2;          // hi half-wave takes K+2,K+3
    const int ncol  = wave * 16 + l15;

    const float* wg = w + (size_t)e * I_ * H_ + hbase;

    for (int ro = 0; ro < cnt; ro += MSUP) {
        const int mrows = min(cnt - ro, MSUP);
        const int nmg   = (mrows + 15) >> 4;   // 1 or 2, uniform per block

        __syncthreads();                        // protect rl/xl vs previous pass
        if (t < MSUP) rl[t] = (ro + t < cnt) ? rowlist[off + ro + t] : -1;
        __syncthreads();

        // ---- stage x rows (full K) once per pass: 32 x 512 floats ----
#pragma unroll
        for (int it = 0; it < (MSUP * I_) / (256 * 4); ++it) {   // 16 iters
            int f   = it * 256 + t;
            int m   = f >> 7;                  // 128 float4 per row
            int cv  = (f & 127) * 4;
            int row = rl[m];
            v4f d = {};
            if (row >= 0) d = *(const v4f*)(x + (size_t)row * I_ + cv);
            *(v4f*)&xl[m * XSTR + cv] = d;
        }
        __syncthreads();

        v8f acc[2] = {v8f{}, v8f{}};
        if (nmg == 1) gemm_pass<1>(wg, wl, xl, t, l15, koff, ncol, acc);
        else          gemm_pass<2>(wg, wl, xl, t, l15, koff, ncol, acc);

        // ---- store D: VGPR j -> (M = j + 8*lhalf, N = l15) ----
        const int gcol = hbase + ncol;
#pragma unroll
        for (int j = 0; j < 8; ++j) {
            int mr = j + 8 * lhalf;
            if (mr < mrows) y[(size_t)rl[mr] * H_ + gcol] = acc[0][j];
        }
        if (nmg > 1) {
#pragma unroll
            for (int j = 0; j < 8; ++j) {
                int mr = 16 + j + 8 * lhalf;
                if (mr < mrows) y[(size_t)rl[mr] * H_ + gcol] = acc[1][j];
            }
        }
    }
}

// ---------------------------------------------------------------------------
// Kernel 3: gate + topk combine: out[t] = ew[t,0]*y[2t] + ew[t,1]*y[2t+1]
// ---------------------------------------------------------------------------
__global__ __launch_bounds__(256) void moe_combine(const float* __restrict__ y,
                                                   const float* __restrict__ ew,
                                                   float* __restrict__ out) {
    size_t i4 = ((size_t)blockIdx.x * 256 + threadIdx.x) * 4;   // over T_*H_ floats
    size_t tt = i4 / H_;
    size_t h  = i4 % H_;
    float w0 = ew[tt * 2 + 0];
    float w1 = ew[tt * 2 + 1];
    v4f y0 = *(const v4f*)(y + (tt * 2 + 0) * (size_t)H_ + h);
    v4f y1 = *(const v4f*)(y + (tt * 2 + 1) * (size_t)H_ + h);
    v4f o  = y0 * w0 + y1 * w1;
    *(v4f*)(out + i4) = o;
}

// ---------------------------------------------------------------------------
extern "C" void kernel_launch(void* const* d_in, const int* in_sizes, int n_in,
                              void* d_out, int out_size, void* d_ws, size_t ws_size,
                              hipStream_t stream) {
    const float* x  = (const float*)d_in[0];   // [R_, I_]
    const float* w  = (const float*)d_in[1];   // [E_, I_, H_]
    const int*   ce = (const int*)d_in[2];     // [T_, K_]
    const float* ew = (const float*)d_in[3];   // [T_, K_]
    float* out = (float*)d_out;                // [T_, H_]

    // workspace layout: y [R_*H_ floats] | counts[E_] | offsets[E_] | rowlist[R_]
    float* y       = (float*)d_ws;
    int*   meta    = (int*)((char*)d_ws + (size_t)R_ * H_ * sizeof(float));
    int*   counts  = meta;
    int*   offsets = meta + E_;
    int*   rowlist = meta + 2 * E_;

    moe_bin<<<1, 256, 0, stream>>>(ce, counts, offsets, rowlist);
    moe_gemm<<<dim3(E_, H_ / HTILE), 256, 0, stream>>>(x, w, counts, offsets, rowlist, y);
    moe_combine<<<(T_ * (size_t)H_) / (256 * 4), 256, 0, stream>>>(y, ew, out);
}